// FastAttention_42099269436266
// MI455X (gfx1250) — compile-verified
//
#include <hip/hip_runtime.h>

// Performer (FAVOR+) attention, MI455X / gfx1250, f32 WMMA path.
// B*H = 64 heads, N = 4096, D = E = 64, M = 266 (padded to 272 = 17 * 16).

#define N_SEQ   4096
#define DDIM    64
#define MFEAT   266
#define MPAD    272
#define NHEADS  64
#define NORMF   0.35355339059327373f   // 64^-0.25
#define RATIO_F 0.061313933948496582f  // 266^-0.5
#define EPS_F   1e-4f

typedef float v2f __attribute__((ext_vector_type(2)));
typedef float v8f __attribute__((ext_vector_type(8)));

__device__ __forceinline__ v8f wmma4(v2f a, v2f b, v8f c) {
  // D = A(16x4,f32) * B(4x16,f32) + C(16x16,f32)
  return __builtin_amdgcn_wmma_f32_16x16x4_f32(false, a, false, b, (short)0, c,
                                               false, false);
}

// Monotonic float <-> uint mapping so global max uses exact integer atomicMax.
__device__ __forceinline__ unsigned f2key(float f) {
  unsigned u = __float_as_uint(f);
  return (u & 0x80000000u) ? ~u : (u | 0x80000000u);
}
__device__ __forceinline__ float key2f(unsigned k) {
  unsigned u = (k & 0x80000000u) ? (k & 0x7fffffffu) : ~k;
  return __uint_as_float(u);
}

// A-fragment from row-major [rows][stride] buffer: element (r,k) at
// lane = r + 16*(k>>1), vgpr = k&1.
__device__ __forceinline__ v2f afrag(const float* buf, int r0, int kk, int stride,
                                     int lrow, int lhalf) {
  const float* p = buf + (r0 + lrow) * stride + kk + (lhalf << 1);
  v2f a; a.x = p[0]; a.y = p[1]; return a;
}

// B-fragment of P^T (B[k][m] = P[m][k]), zero for padded m >= 266.
__device__ __forceinline__ v2f bfragP(const float* __restrict__ P, int m0, int kk,
                                      int lrow, int lhalf) {
  int m = m0 + lrow;
  v2f b;
  if (m < MFEAT) {
    const float* p = P + m * DDIM + kk + (lhalf << 1);
    b.x = p[0]; b.y = p[1];
  } else { b.x = 0.f; b.y = 0.f; }
  return b;
}

// B-fragment from row-major [K][stride] buffer (rows are the K dim).
__device__ __forceinline__ v2f bfragK(const float* buf, int kk, int c0, int stride,
                                      int lrow, int lhalf) {
  int k = kk + (lhalf << 1);
  v2f b;
  b.x = buf[k * stride + c0 + lrow];
  b.y = buf[(k + 1) * stride + c0 + lrow];
  return b;
}

__global__ void perf_init_kernel(unsigned* kmax_key) { *kmax_key = 0u; }

// ---------------------------------------------------------------------------
// Kernel A: global max of k_dash = (NORMF*k) @ P^T   (pre-exp stabilizer)
// grid = 64 heads * 32 chunks, block = 256 (8 waves, 16 rows each)
// ---------------------------------------------------------------------------
__global__ __launch_bounds__(256) void perf_kmax_kernel(
    const float* __restrict__ K, const float* __restrict__ P,
    unsigned* __restrict__ kmax_key) {
  extern __shared__ float sm[];
  float* Pl  = sm;                 // 272*64 (zero-padded)
  float* Ks  = Pl + MPAD * DDIM;   // 128*64 (scaled rows)
  float* red = Ks + 128 * DDIM;    // 8
  int tid = threadIdx.x, bx = blockIdx.x;
  int head = bx >> 5;
  int n0   = (bx & 31) * 128;

  {
    const float4* P4 = (const float4*)P;
    float4* Pl4 = (float4*)Pl;
    for (int i = tid; i < MPAD * DDIM / 4; i += 256) {
      float4 t = {0.f, 0.f, 0.f, 0.f};
      if (i < MFEAT * DDIM / 4) t = P4[i];
      Pl4[i] = t;
    }
    const float4* kg4 =
        (const float4*)(K + ((size_t)head * N_SEQ + n0) * DDIM);
    float4* Ks4 = (float4*)Ks;
    for (int i = tid; i < 128 * DDIM / 4; i += 256) {
      float4 t = kg4[i];
      t.x *= NORMF; t.y *= NORMF; t.z *= NORMF; t.w *= NORMF;
      Ks4[i] = t;
    }
  }
  __syncthreads();

  int lane = tid & 31, w = tid >> 5;
  int lrow = lane & 15, lhalf = lane >> 4;

  v2f af[16];
#pragma unroll
  for (int s = 0; s < 16; ++s) af[s] = afrag(Ks, w * 16, 4 * s, DDIM, lrow, lhalf);

  float wmax = -3.402823466e38f;
  for (int mt = 0; mt < 17; ++mt) {
    int m0 = mt * 16;
    v2f bf[16];
#pragma unroll
    for (int s = 0; s < 16; ++s)
      bf[s] = afrag(Pl, m0, 4 * s, DDIM, lrow, lhalf);  // B[k][m] = Pl[m][k]
    v8f acc0 = {}, acc1 = {};
#pragma unroll
    for (int s = 0; s < 16; s += 2) {
      acc0 = wmma4(af[s], bf[s], acc0);
      acc1 = wmma4(af[s + 1], bf[s + 1], acc1);
    }
    if ((mt < 16) || (lrow < MFEAT - 256))
#pragma unroll
      for (int j = 0; j < 8; ++j) wmax = fmaxf(wmax, acc0[j] + acc1[j]);
  }
  for (int off = 16; off; off >>= 1)
    wmax = fmaxf(wmax, __shfl_xor(wmax, off, 32));
  if (lane == 0) red[w] = wmax;
  __syncthreads();
  if (tid == 0) {
    float m = red[0];
    for (int i = 1; i < 8; ++i) m = fmaxf(m, red[i]);
    atomicMax(kmax_key, f2key(m));
  }
}

// ---------------------------------------------------------------------------
// Kernel B: per head: k' = ratio*(exp(kdash - diag - kmax) + eps),
//           k_sum[m] = sum_n k'[n][m],  context[m][e] = sum_n k'[n][m]*v[n][e]
// grid = 64 heads, block = 544 (17 waves; wave w owns m-tile w)
// ---------------------------------------------------------------------------
__global__ __launch_bounds__(544) void perf_ctx_kernel(
    const float* __restrict__ K, const float* __restrict__ V,
    const float* __restrict__ P, const unsigned* __restrict__ kmax_key,
    float* __restrict__ ws_ksum, float* __restrict__ ws_ctx) {
  extern __shared__ float sm[];
  float* Ks = sm;                  // 64*64 scaled k rows
  float* Vs = Ks + 64 * DDIM;      // 64*64 v rows
  float* Kp = Vs + 64 * DDIM;      // 64*272 k' rows
  float* Dg = Kp + 64 * MPAD;      // 64 diag
  int tid = threadIdx.x, head = blockIdx.x;
  int lane = tid & 31, w = tid >> 5;         // w in [0,17)
  int lrow = lane & 15, lhalf = lane >> 4;
  float kmax = key2f(*kmax_key);
  const float4* kg4 = (const float4*)(K + (size_t)head * N_SEQ * DDIM);
  const float4* vg4 = (const float4*)(V + (size_t)head * N_SEQ * DDIM);

  int m0 = w * 16;
  int mcol = m0 + lrow;
  bool mval = mcol < MFEAT;

  // P column-tile fragments for this wave are invariant: load once.
  v2f pb[16];
#pragma unroll
  for (int s = 0; s < 16; ++s) pb[s] = bfragP(P, m0, 4 * s, lrow, lhalf);

  v8f ctx[4] = {};
  float ksacc = 0.f;

  for (int n0 = 0; n0 < N_SEQ; n0 += 64) {
    __syncthreads();  // previous iteration's reads of Ks/Vs/Kp are done
    {
      float4* Ks4 = (float4*)Ks;
      float4* Vs4 = (float4*)Vs;
      int base = n0 * DDIM / 4;
      for (int i = tid; i < 64 * DDIM / 4; i += 544) {
        float4 t = kg4[base + i];
        t.x *= NORMF; t.y *= NORMF; t.z *= NORMF; t.w *= NORMF;
        Ks4[i] = t;
        Vs4[i] = vg4[base + i];
      }
    }
    __syncthreads();
    if (tid < 64) {  // diag[r] = 0.5 * sum_d (norm*k)^2, rotated for banks
      float s = 0.f;
      for (int i = 0; i < DDIM; ++i) {
        float x = Ks[tid * DDIM + ((i + tid) & 63)];
        s += x * x;
      }
      Dg[tid] = 0.5f * s;
    }
    __syncthreads();
    // phase 1: wave w computes k' col-tile m0 for 4 row tiles
    for (int rt = 0; rt < 4; ++rt) {
      v2f af[16];
#pragma unroll
      for (int s = 0; s < 16; ++s)
        af[s] = afrag(Ks, rt * 16, 4 * s, DDIM, lrow, lhalf);
      v8f acc0 = {}, acc1 = {};
#pragma unroll
      for (int s = 0; s < 16; s += 2) {
        acc0 = wmma4(af[s], pb[s], acc0);
        acc1 = wmma4(af[s + 1], pb[s + 1], acc1);
      }
#pragma unroll
      for (int j = 0; j < 8; ++j) {
        int r = rt * 16 + j + 8 * lhalf;
        float dd = acc0[j] + acc1[j];
        float val = mval ? RATIO_F * (expf(dd - Dg[r] - kmax) + EPS_F) : 0.f;
        Kp[r * MPAD + mcol] = val;
        ksacc += val;
      }
    }
    __syncthreads();
    // phase 2: ctx[m-tile w] += k'^T @ v; shared A-group feeds 4 e-chains
    for (int nk = 0; nk < 64; nk += 16) {
      v2f af[4];
#pragma unroll
      for (int s = 0; s < 4; ++s)
        af[s] = bfragK(Kp, nk + 4 * s, m0, MPAD, lrow, lhalf);  // A[m][n]
#pragma unroll
      for (int et = 0; et < 4; ++et) {
        v2f bf[4];
#pragma unroll
        for (int s = 0; s < 4; ++s)
          bf[s] = bfragK(Vs, nk + 4 * s, et * 16, DDIM, lrow, lhalf);
#pragma unroll
        for (int s = 0; s < 4; ++s) ctx[et] = wmma4(af[s], bf[s], ctx[et]);
      }
    }
  }
  ksacc += __shfl_down(ksacc, 16, 32);
  if (lane < 16) ws_ksum[head * MPAD + m0 + lane] = ksacc;
  float* cg = ws_ctx + (size_t)head * MPAD * DDIM;
#pragma unroll
  for (int et = 0; et < 4; ++et)
#pragma unroll
    for (int j = 0; j < 8; ++j) {
      int m = m0 + j + 8 * lhalf;
      cg[m * DDIM + et * 16 + lrow] = ctx[et][j];
    }
}

// ---------------------------------------------------------------------------
// Kernel C: q' (row-max stabilizer), D_inv, out = (q' @ ctx) * D_inv
// grid = 64 heads * 64 chunks, block = 128 (4 waves, 16 rows each)
// ---------------------------------------------------------------------------
__global__ __launch_bounds__(128) void perf_out_kernel(
    const float* __restrict__ Q, const float* __restrict__ P,
    const float* __restrict__ ws_ksum, const float* __restrict__ ws_ctx,
    float* __restrict__ Out) {
  extern __shared__ float sm[];
  float* Cx = sm;                  // 272*64 context
  float* Su = Cx + MPAD * DDIM;    // 272 k_sum
  float* Qs = Su + MPAD;           // 64*64 scaled q rows
  float* Qp = Qs + 64 * DDIM;      // 64*272 dd then q'
  float* Dv = Qp + 64 * MPAD;      // 64 D_inv
  int tid = threadIdx.x, bx = blockIdx.x;
  int head = bx >> 6;
  int n0   = (bx & 63) * 64;
  int lane = tid & 31, w = tid >> 5;
  int lrow = lane & 15, lhalf = lane >> 4;

  {
    const float4* cxg4 = (const float4*)(ws_ctx + (size_t)head * MPAD * DDIM);
    float4* Cx4 = (float4*)Cx;
    for (int i = tid; i < MPAD * DDIM / 4; i += 128) Cx4[i] = cxg4[i];
    const float4* su4 = (const float4*)(ws_ksum + head * MPAD);
    float4* Su4 = (float4*)Su;
    for (int i = tid; i < MPAD / 4; i += 128) Su4[i] = su4[i];
    const float4* qg4 = (const float4*)(Q + ((size_t)head * N_SEQ + n0) * DDIM);
    float4* Qs4 = (float4*)Qs;
    for (int i = tid; i < 64 * DDIM / 4; i += 128) {
      float4 t = qg4[i];
      t.x *= NORMF; t.y *= NORMF; t.z *= NORMF; t.w *= NORMF;
      Qs4[i] = t;
    }
  }
  __syncthreads();

  int r = w * 16 + lrow;  // this lane-pair's row
  float dsum = 0.f;       // diag in registers: half the d-range per lane half
  for (int i = 0; i < 32; ++i) {
    float x = Qs[r * DDIM + lhalf * 32 + i];
    dsum += x * x;
  }
  dsum += __shfl_xor(dsum, 16, 32);
  float diag = 0.5f * dsum;

  // phase 1: dd = q_scaled @ P^T into Qp (raw, pre-exp)
  v2f qa[16];
#pragma unroll
  for (int s = 0; s < 16; ++s) qa[s] = afrag(Qs, w * 16, 4 * s, DDIM, lrow, lhalf);
  for (int mt = 0; mt < 17; ++mt) {
    int m0 = mt * 16;
    v2f pb[16];
#pragma unroll
    for (int s = 0; s < 16; ++s) pb[s] = bfragP(P, m0, 4 * s, lrow, lhalf);
    v8f acc0 = {}, acc1 = {};
#pragma unroll
    for (int s = 0; s < 16; s += 2) {
      acc0 = wmma4(qa[s], pb[s], acc0);
      acc1 = wmma4(qa[s + 1], pb[s + 1], acc1);
    }
#pragma unroll
    for (int j = 0; j < 8; ++j)
      Qp[(w * 16 + j + 8 * lhalf) * MPAD + m0 + lrow] = acc0[j] + acc1[j];
  }
  __syncthreads();

  // per-row max over valid cols, then exp + D = q'.k_sum
  float rmax = -3.402823466e38f;
  {
    const float* row = Qp + r * MPAD;
    int c0 = lhalf * 133;               // 266 = 2*133
    for (int c = c0; c < c0 + 133; ++c) rmax = fmaxf(rmax, row[c]);
    rmax = fmaxf(rmax, __shfl_xor(rmax, 16, 32));
  }
  {
    float* row = Qp + r * MPAD;
    float dsumD = 0.f;
    int c0 = lhalf * 136;               // 272 = 2*136
    for (int c = c0; c < c0 + 136; ++c) {
      float val = (c < MFEAT)
                      ? RATIO_F * (expf(row[c] - diag - rmax) + EPS_F)
                      : 0.f;
      row[c] = val;
      dsumD += val * Su[c];
    }
    dsumD += __shfl_xor(dsumD, 16, 32);
    if (lhalf == 0) Dv[r] = 1.0f / dsumD;
  }
  __syncthreads();

  // phase 2: out = q'(16x272) @ ctx(272x64); shared A-group feeds 4 e-chains
  v8f acc[4] = {};
  for (int kk = 0; kk < MPAD; kk += 16) {
    v2f af[4];
#pragma unroll
    for (int s = 0; s < 4; ++s)
      af[s] = afrag(Qp, w * 16, kk + 4 * s, MPAD, lrow, lhalf);
#pragma unroll
    for (int et = 0; et < 4; ++et) {
      v2f bf[4];
#pragma unroll
      for (int s = 0; s < 4; ++s)
        bf[s] = bfragK(Cx, kk + 4 * s, et * 16, DDIM, lrow, lhalf);
#pragma unroll
      for (int s = 0; s < 4; ++s) acc[et] = wmma4(af[s], bf[s], acc[et]);
    }
  }
#pragma unroll
  for (int et = 0; et < 4; ++et) {
    int e0 = et * 16;
#pragma unroll
    for (int j = 0; j < 8; ++j) {
      int rr = w * 16 + j + 8 * lhalf;
      Out[((size_t)head * N_SEQ + n0 + rr) * DDIM + e0 + lrow] =
          acc[et][j] * Dv[rr];
    }
  }
}

extern "C" void kernel_launch(void* const* d_in, const int* in_sizes, int n_in,
                              void* d_out, int out_size, void* d_ws,
                              size_t ws_size, hipStream_t stream) {
  (void)in_sizes; (void)n_in; (void)out_size; (void)ws_size;
  const float* q = (const float*)d_in[0];
  const float* k = (const float*)d_in[1];
  const float* v = (const float*)d_in[2];
  const float* P = (const float*)d_in[3];
  float* out = (float*)d_out;

  unsigned* kmax_key = (unsigned*)d_ws;
  float* ws_ksum = (float*)d_ws + 64;                 // 64 heads * 272
  float* ws_ctx  = ws_ksum + NHEADS * MPAD;           // 64 heads * 272 * 64

  perf_init_kernel<<<dim3(1), dim3(1), 0, stream>>>(kmax_key);

  size_t smA = (size_t)(MPAD * DDIM + 128 * DDIM + 8) * sizeof(float);
  perf_kmax_kernel<<<dim3(NHEADS * 32), dim3(256), smA, stream>>>(k, P,
                                                                  kmax_key);

  size_t smB = (size_t)(64 * DDIM + 64 * DDIM + 64 * MPAD + 64) * sizeof(float);
  perf_ctx_kernel<<<dim3(NHEADS), dim3(544), smB, stream>>>(k, v, P, kmax_key,
                                                            ws_ksum, ws_ctx);

  size_t smC =
      (size_t)(MPAD * DDIM + MPAD + 64 * DDIM + 64 * MPAD + 64) * sizeof(float);
  perf_out_kernel<<<dim3(NHEADS * 64), dim3(128), smC, stream>>>(
      q, P, ws_ksum, ws_ctx, out);
}